// VQ_5935644803109
// MI455X (gfx1250) — compile-verified
//
#include <hip/hip_runtime.h>

// ---------------------------------------------------------------------------
// VQ nearest-codebook-entry kernel for MI455X (gfx1250, wave32, WMMA + async
// global->LDS staging).
//   x:     [32, 64, 64, 64] f32 NCHW  (C = D = 64 embedding dim)
//   embed: [512, 64] f32 codebook
// d_out = quantized [B,H,W,C] f32 (8,388,608) ++ indices [B,H,W] (131,072,
// stored as float).
//
// Roofline: HBM floor = 32MB x-read + 32MB quantized-write ~ 2.8us @23.3TB/s.
// Naive per-wave codebook re-reads would need ~350TB/s from L2, so each
// 256-thread workgroup stages the 128KB codebook into LDS ONCE via
// global_load_async_to_lds_b128 (ASYNCcnt), computes e_sq locally, and runs
// the [16-row x 512-code] distance GEMM with V_WMMA_F32_16X16X4_F32, feeding
// the A operand from LDS (ds_load_b64). x_sq is a per-row constant and is
// dropped from the argmin: proxy = e_sq[k] - 2*<x,e_k>.
// ---------------------------------------------------------------------------

typedef float v2f __attribute__((ext_vector_type(2)));
typedef float v4f __attribute__((ext_vector_type(4)));
typedef float v8f __attribute__((ext_vector_type(8)));

#define K_CODES 512
#define D_DIM   64
#define HW      4096        // H*W
#define CHW     262144      // C*H*W
#define N_ROWS  131072      // B*H*W
#define LDS_BYTES ((K_CODES * D_DIM + K_CODES) * sizeof(float))  // 130KB

__global__ __launch_bounds__(256) void vq_main_kernel(
    const float* __restrict__ x,
    const float* __restrict__ embed,
    float* __restrict__ outq,     // quantized, NHWC, row n at outq + n*64
    float* __restrict__ outi)     // indices as float
{
    extern __shared__ float smem[];
    float* sh_embed = smem;                      // [512][64] = 128KB
    float* sh_esq   = smem + K_CODES * D_DIM;    // [512]     = 2KB

    const int tid  = threadIdx.x;
    const int lane = tid & 31;
    const int wave = tid >> 5;
    const int r    = lane & 15;     // x-row within the wave's 16-row tile
    const int half = lane >> 4;     // K-slice / code-half selector

    // ---- Stage codebook to LDS with async global->LDS B128 copies --------
    // 8192 x 16B transfers; 256 threads x 32 iterations, one lane-transfer
    // each. Tracked by ASYNCcnt, drained with s_wait_asynccnt.
#pragma unroll
    for (int j = 0; j < 32; ++j) {
        const int idx4 = (tid + 256 * j) * 4;               // float index
        const float* g = embed + idx4;
        const unsigned l = (unsigned)(uintptr_t)(sh_embed + idx4); // LDS addr
        asm volatile("global_load_async_to_lds_b128 %0, %1, off"
                     :: "v"(l), "v"(g) : "memory");
    }
    asm volatile("s_wait_asynccnt 0x0" ::: "memory");
    __syncthreads();

    // ---- e_sq[k] = ||e_k||^2 from LDS-resident codebook ------------------
    for (int k = tid; k < K_CODES; k += 256) {
        const float* row = sh_embed + k * D_DIM;
        float s = 0.0f;
#pragma unroll
        for (int j = 0; j < D_DIM; ++j) s = fmaf(row[j], row[j], s);
        sh_esq[k] = s;
    }
    __syncthreads();

    // ---- Load B operand (x features) once; reused for all 32 code tiles --
    // V_WMMA_F32_16X16X4_F32 B layout: lanes 0-15 hold N=0..15, K={4s,4s+1};
    // lanes 16-31 hold K={4s+2,4s+3}. NCHW => feature stride HW=4096.
    const int rowBase = (blockIdx.x * 8 + wave) * 16;
    const int n       = rowBase + r;
    const long xbase  = (long)(n >> 12) * CHW + (n & 4095);

    v2f bx[16];
#pragma unroll
    for (int s = 0; s < 16; ++s) {
        const int d0 = 4 * s + 2 * half;
        bx[s].x = x[xbase + (long)d0 * HW];
        bx[s].y = x[xbase + (long)(d0 + 1) * HW];
    }

    float bestVal = __builtin_inff();
    int   bestIdx = 0;

    for (int t = 0; t < 32; ++t) {
        const int codeBase = t * 16;

        // A operand (codebook rows) from LDS: lane's code row is
        // codeBase + (lane%16), contiguous float2 at feature 4s + 2*half.
        const float* arow = sh_embed + (codeBase + r) * D_DIM + 2 * half;
        v2f a[16];
#pragma unroll
        for (int s = 0; s < 16; ++s)
            a[s] = *(const v2f*)(arow + 4 * s);             // ds_load_b64

        v8f c = {0.f, 0.f, 0.f, 0.f, 0.f, 0.f, 0.f, 0.f};
#pragma unroll
        for (int s = 0; s < 16; ++s)
            c = __builtin_amdgcn_wmma_f32_16x16x4_f32(
                    /*neg_a=*/false, a[s], /*neg_b=*/false, bx[s],
                    /*c_mod=*/(short)0, c, /*reuse_a=*/false, /*reuse_b=*/false);

        // C layout: this lane sees x-row N=r, codes M = 8*half + v.
        const float* ep = sh_esq + codeBase + 8 * half;
        v4f e0 = *(const v4f*)(ep);
        v4f e1 = *(const v4f*)(ep + 4);
#pragma unroll
        for (int v = 0; v < 8; ++v) {
            const float ev    = (v < 4) ? e0[v] : e1[v - 4];
            const float proxy = fmaf(-2.0f, c[v], ev);
            const int   k     = codeBase + 8 * half + v;
            if (proxy < bestVal) { bestVal = proxy; bestIdx = k; }
        }
    }

    // ---- Combine the two code-halves of row r (lanes L and L^16) ---------
    // Tie-break on smaller index to match jnp.argmin (first occurrence).
    const float oVal = __shfl_xor(bestVal, 16, 32);
    const int   oIdx = __shfl_xor(bestIdx, 16, 32);
    if (oVal < bestVal || (oVal == bestVal && oIdx < bestIdx)) {
        bestVal = oVal; bestIdx = oIdx;
    }

    if (half == 0) outi[n] = (float)bestIdx;

    // ---- Gather winning code row from LDS; lane pair splits it 32/32 -----
    const v4f* src = (const v4f*)(sh_embed + bestIdx * D_DIM + 32 * half);
    v4f*       dst = (v4f*)(outq + (long)n * D_DIM + 32 * half);
#pragma unroll
    for (int j = 0; j < 8; ++j) dst[j] = src[j];
}

extern "C" void kernel_launch(void* const* d_in, const int* in_sizes, int n_in,
                              void* d_out, int out_size, void* d_ws, size_t ws_size,
                              hipStream_t stream) {
    const float* x     = (const float*)d_in[0];   // 8,388,608 f32
    const float* embed = (const float*)d_in[1];   // 32,768 f32
    float* outq = (float*)d_out;                  // quantized block
    float* outi = outq + (size_t)N_ROWS * D_DIM;  // indices block

    // 131072 rows / (8 waves * 16 rows) = 1024 blocks of 256 threads.
    // 130KB dynamic LDS per workgroup -> 2 workgroups per 320KB WGP.
    vq_main_kernel<<<1024, 256, LDS_BYTES, stream>>>(x, embed, outq, outi);
}